// MeshPoolFace_7584912245099
// MI455X (gfx1250) — compile-verified
//
#include <hip/hip_runtime.h>
#include <stdint.h>

// Problem dims (from reference)
#define B_DIM 16
#define C_DIM 256
#define F_DIM 16000
#define T_DIM 8000

#define TPB   256
#define FT    1024                       // faces per tile (4 per thread, float4)
#define NFT   ((F_DIM + FT - 1) / FT)    // 16 face tiles
#define CPB   64                         // channels per block
#define CGRID (C_DIM / CPB)              // 4

// CDNA5 async global->LDS copy (GVS mode: SGPR64 base + VGPR32 byte offset).
// First operand = LDS byte-offset VGPR, tracked on ASYNCcnt.
__device__ __forceinline__ void async_b128_to_lds(uint32_t lds_off, uint32_t goff,
                                                  const void* sbase) {
  asm volatile("global_load_async_to_lds_b128 %0, %1, %2"
               :: "v"(lds_off), "v"(goff), "s"(sbase)
               : "memory");
}

__global__ void __launch_bounds__(TPB)
count_faces(const int* __restrict__ gid, int* __restrict__ cnt) {
  int i = blockIdx.x * blockDim.x + threadIdx.x;        // over B*F
  if (i < B_DIM * F_DIM) {
    int b = i / F_DIM;
    atomicAdd(&cnt[b * T_DIM + gid[i]], 1);
  }
}

__global__ void __launch_bounds__(TPB)
make_inv(const int* __restrict__ cnt, float* __restrict__ inv) {
  int i = blockIdx.x * blockDim.x + threadIdx.x;        // over B*TARGET
  if (i < B_DIM * T_DIM) {
    int c = cnt[i];
    inv[i] = 1.0f / (float)(c > 0 ? c : 1);
  }
}

// Fused weighted scatter: out[b][c][gid] += fe[b][c][f] * inv[b][gid].
// gridDim = (B*NFT, C/CPB); each thread owns 4 consecutive faces and loops
// CPB channels, double-buffering fe through LDS via async copies.
__global__ void __launch_bounds__(TPB)
scatter_mean(const float* __restrict__ fe, const int* __restrict__ gid,
             const float* __restrict__ inv, float* __restrict__ out) {
  __shared__ __align__(16) float buf[2][FT];

  const int t  = threadIdx.x;
  const int b  = blockIdx.x / NFT;
  const int f0 = (blockIdx.x % NFT) * FT;
  const int c0 = blockIdx.y * CPB;
  const int f  = f0 + t * 4;
  if (f >= F_DIM) return;   // F_DIM % 4 == 0 -> whole float4 valid; wave-aligned cut

  // Per-thread group ids + mean weights, reused across all CPB channels.
  const int4 g4 = *reinterpret_cast<const int4*>(gid + b * F_DIM + f);
  const float* invb = inv + b * T_DIM;
  const float w0 = invb[g4.x], w1 = invb[g4.y], w2 = invb[g4.z], w3 = invb[g4.w];

  float* outb = out + ((size_t)b * C_DIM + c0) * (size_t)T_DIM;
  float* p0 = outb + g4.x;
  float* p1 = outb + g4.y;
  float* p2 = outb + g4.z;
  float* p3 = outb + g4.w;

  const void* fe_b = (const void*)(fe + (size_t)b * C_DIM * F_DIM);
  uint32_t goff = ((uint32_t)c0 * (uint32_t)F_DIM + (uint32_t)f) * 4u;  // < 2^31
  // LDS byte offsets: each lane stages exactly the 16B it will read back.
  const uint32_t lds0 = (uint32_t)(uintptr_t)(&buf[0][t * 4]);
  const uint32_t lds1 = (uint32_t)(uintptr_t)(&buf[1][t * 4]);

  // Prologue: stage channel c0 into buffer 0.
  async_b128_to_lds(lds0, goff, fe_b);
  goff += (uint32_t)F_DIM * 4u;

  for (int cc = 0; cc < CPB; ++cc) {
    if (cc + 1 < CPB) {
      // Stage next channel into the other buffer, then wait for current one.
      async_b128_to_lds((cc & 1) ? lds0 : lds1, goff, fe_b);
      goff += (uint32_t)F_DIM * 4u;
      asm volatile("s_wait_asynccnt 1" ::: "memory");
    } else {
      asm volatile("s_wait_asynccnt 0" ::: "memory");
    }
    const float4 v = *reinterpret_cast<const float4*>(&buf[cc & 1][t * 4]);
    // Non-returning global_atomic_add_f32 (STOREcnt): fire-and-forget.
    atomicAdd(p0, v.x * w0);
    atomicAdd(p1, v.y * w1);
    atomicAdd(p2, v.z * w2);
    atomicAdd(p3, v.w * w3);
    p0 += T_DIM; p1 += T_DIM; p2 += T_DIM; p3 += T_DIM;
  }
}

extern "C" void kernel_launch(void* const* d_in, const int* in_sizes, int n_in,
                              void* d_out, int out_size, void* d_ws, size_t ws_size,
                              hipStream_t stream) {
  (void)in_sizes; (void)n_in; (void)ws_size;
  const float* fe  = (const float*)d_in[0];
  const int*   gid = (const int*)d_in[1];   // int32 per harness convention
  float* out = (float*)d_out;

  int*   cnt = (int*)d_ws;                                           // B*T ints
  float* inv = (float*)((char*)d_ws + (size_t)B_DIM * T_DIM * 4);    // B*T floats

  hipMemsetAsync(out, 0, (size_t)out_size * sizeof(float), stream);
  hipMemsetAsync(cnt, 0, (size_t)B_DIM * T_DIM * sizeof(int), stream);

  count_faces<<<(B_DIM * F_DIM + TPB - 1) / TPB, TPB, 0, stream>>>(gid, cnt);
  make_inv<<<(B_DIM * T_DIM + TPB - 1) / TPB, TPB, 0, stream>>>(cnt, inv);

  dim3 grid(B_DIM * NFT, CGRID, 1);
  scatter_mean<<<grid, TPB, 0, stream>>>(fe, gid, inv, out);
}